// RelationEncoding_79860621902601
// MI455X (gfx1250) — compile-verified
//
#include <hip/hip_runtime.h>
#include <cstdint>

typedef __attribute__((ext_vector_type(2))) float v2f;
typedef __attribute__((ext_vector_type(8))) float v8f;

#define B_    4
#define N_    4096
#define NS    32
#define ROWS  (B_*N_*NS)      /* 524288 flattened (b,n,s) rows */
#define TILES (ROWS/16)       /* 32768 row tiles */
#define GRID_L 512            /* layer-kernel workgroups; 512*8 waves = 4096 -> 8 tiles/wave */
#define R2    0.04f
#define NEG_SLOPE_ 0.2f
#define EPS_  1e-5f

__device__ __forceinline__ float leaky(float x) { return fmaxf(x, NEG_SLOPE_ * x); }

// ---------------------------------------------------------------------------
// Kernel 1: ball query. One batch's xyz (48KB) lives in LDS; each thread scans
// all 4096 points in ascending index order, keeping the first 32 within radius.
// ---------------------------------------------------------------------------
__global__ void ball_query_kernel(const float* __restrict__ points, int* __restrict__ idx)
{
    __shared__ float sx[N_], sy[N_], sz[N_];
    const int b   = blockIdx.x >> 4;   // 16 blocks of 256 query points per batch
    const int blk = blockIdx.x & 15;
    const float* P = points + (size_t)b * 3 * N_;
    for (int i = threadIdx.x; i < N_; i += 256) {
        sx[i] = P[i]; sy[i] = P[N_ + i]; sz[i] = P[2*N_ + i];
    }
    __syncthreads();
    const int n = blk * 256 + threadIdx.x;
    const float qx = sx[n], qy = sy[n], qz = sz[n];
    int* out = idx + (((size_t)b * N_ + n) << 5);
    int cnt = 0, first = n;
    for (int j = 0; j < N_; ++j) {
        float dx = sx[j] - qx, dy = sy[j] - qy, dz = sz[j] - qz;
        float d2 = dx*dx + dy*dy + dz*dz;
        if (d2 <= R2) {
            if (cnt == 0) first = j;
            out[cnt] = j;
            if (++cnt == NS) break;
        }
    }
    for (; cnt < NS; ++cnt) out[cnt] = first;
}

// ---------------------------------------------------------------------------
// Kernel 2: layer0. Builds 6ch edge features on the fly (K padded to 8),
// 2x V_WMMA_F32_16X16X4_F32 per 16-row x 16-ch tile, fused per-channel
// sum/sum^2 partials (deterministic workgroup-level tree reduction).
// ---------------------------------------------------------------------------
__global__ void layer0_kernel(const float* __restrict__ points,
                              const int*   __restrict__ idx,
                              const float* __restrict__ W0,    // 32 x 6
                              const float* __restrict__ Bs0,   // 32
                              float*       __restrict__ bufOut,// ROWS x 32
                              float*       __restrict__ partial)// [GRID_L][128]
{
    __shared__ float s_w[32 * 8];
    __shared__ float s_bias[32];
    __shared__ float s_red[8][64][2];
    for (int i = threadIdx.x; i < 32 * 8; i += 256) {
        int o = i >> 3, c = i & 7;
        s_w[i] = (c < 6) ? W0[o * 6 + c] : 0.0f;   // zero-pad K 6 -> 8
    }
    if (threadIdx.x < 32) s_bias[threadIdx.x] = Bs0[threadIdx.x];
    __syncthreads();

    const int lane = threadIdx.x & 31;
    const int wave = threadIdx.x >> 5;
    const int half = lane >> 4;
    const int l15  = lane & 15;

    float ps[2]  = {0.f, 0.f};
    float ps2[2] = {0.f, 0.f};

    for (int t = blockIdx.x * 8 + wave; t < TILES; t += GRID_L * 8) {
        const int row = t * 16 + l15;
        const int b   = row >> 17;            // N_*NS = 2^17 rows per batch
        const int n   = (row >> 5) & (N_ - 1);
        const int gi  = idx[row];
        const float* P = points + (size_t)b * 3 * N_;
        // features: f0..2 = center xyz, f3..5 = grouped - center, f6..7 = 0
        float a0x, a0y, a1x, a1y;
        if (half == 0) {          // K = 0,1 (step0) and 4,5 (step1)
            float cx = P[n], cy = P[N_ + n], cz = P[2*N_ + n];
            a0x = cx; a0y = cy;
            a1x = P[N_ + gi] - cy; a1y = P[2*N_ + gi] - cz;
        } else {                  // K = 2,3 (step0) and 6,7 (step1, zero pad)
            float cx = P[n], cz = P[2*N_ + n];
            a0x = cz; a0y = P[gi] - cx;
            a1x = 0.0f; a1y = 0.0f;
        }
        v2f A0; A0.x = a0x; A0.y = a0y;
        v2f A1; A1.x = a1x; A1.y = a1y;

        v8f acc[2];
        #pragma unroll
        for (int ct = 0; ct < 2; ++ct) {
            float bv = s_bias[ct * 16 + l15];
            #pragma unroll
            for (int v = 0; v < 8; ++v) acc[ct][v] = bv;
        }
        #pragma unroll
        for (int ct = 0; ct < 2; ++ct) {
            const int o  = ct * 16 + l15;
            const int c0 = 2 * half;
            v2f Bf0; Bf0.x = s_w[o*8 + c0];     Bf0.y = s_w[o*8 + c0 + 1];
            v2f Bf1; Bf1.x = s_w[o*8 + 4 + c0]; Bf1.y = s_w[o*8 + 4 + c0 + 1];
            acc[ct] = __builtin_amdgcn_wmma_f32_16x16x4_f32(false, A0, false, Bf0,
                                                            (short)0, acc[ct], false, false);
            acc[ct] = __builtin_amdgcn_wmma_f32_16x16x4_f32(false, A1, false, Bf1,
                                                            (short)0, acc[ct], false, false);
        }
        const int rb = t * 16;
        #pragma unroll
        for (int ct = 0; ct < 2; ++ct) {
            const int ch = ct * 16 + l15;
            #pragma unroll
            for (int v = 0; v < 8; ++v) {
                float d = acc[ct][v];
                bufOut[(size_t)(rb + v + 8*half) * 32 + ch] = d;
                ps[ct]  += d;
                ps2[ct] += d * d;
            }
        }
    }
    #pragma unroll
    for (int ct = 0; ct < 2; ++ct) {
        ps[ct]  += __shfl_xor(ps[ct],  16, 32);
        ps2[ct] += __shfl_xor(ps2[ct], 16, 32);
        if (half == 0) {
            s_red[wave][ct*16 + l15][0] = ps[ct];
            s_red[wave][ct*16 + l15][1] = ps2[ct];
        }
    }
    __syncthreads();
    if (threadIdx.x < 64) {   // 32 channels * {sum, sumsq}
        int ch = threadIdx.x >> 1, j = threadIdx.x & 1;
        float s = 0.f;
        #pragma unroll
        for (int w = 0; w < 8; ++w) s += s_red[w][ch][j];
        partial[(size_t)blockIdx.x * 128 + ch*2 + j] = s;
    }
}

// ---------------------------------------------------------------------------
// Generic 32-in layer: applies previous layer's batchnorm affine + leaky ReLU
// while loading A fragments, chains 8 f32 WMMAs over K=32 per output tile.
// Safe in-place (bufIn == bufOut): each wave owns its 16 rows exclusively.
// ---------------------------------------------------------------------------
template <int OC>
__global__ void layer_ic32_kernel(const float* __restrict__ bufIn,  // ROWS x 32
                                  float*       __restrict__ bufOut, // ROWS x OC
                                  const float* __restrict__ W,      // OC x 32
                                  const float* __restrict__ Bs,     // OC
                                  const float* __restrict__ nrm,    // 32 x {scale,shift}
                                  float*       __restrict__ partial)
{
    constexpr int NCT = OC / 16;
    __shared__ float s_w[OC * 32];
    __shared__ float s_bias[OC];
    __shared__ float s_sc[32], s_sh[32];
    __shared__ float s_red[8][64][2];
    for (int i = threadIdx.x; i < OC * 32; i += 256) s_w[i] = W[i];
    if (threadIdx.x < OC) s_bias[threadIdx.x] = Bs[threadIdx.x];
    if (threadIdx.x < 32) {
        s_sc[threadIdx.x] = nrm[2*threadIdx.x];
        s_sh[threadIdx.x] = nrm[2*threadIdx.x + 1];
    }
    __syncthreads();

    const int lane = threadIdx.x & 31;
    const int wave = threadIdx.x >> 5;
    const int half = lane >> 4;
    const int l15  = lane & 15;

    float ps[NCT], ps2[NCT];
    #pragma unroll
    for (int ct = 0; ct < NCT; ++ct) { ps[ct] = 0.f; ps2[ct] = 0.f; }

    for (int t = blockIdx.x * 8 + wave; t < TILES; t += GRID_L * 8) {
        const int row = t * 16 + l15;
        const float* rp = bufIn + (size_t)row * 32;

        v8f acc[NCT];
        #pragma unroll
        for (int ct = 0; ct < NCT; ++ct) {
            float bv = s_bias[ct * 16 + l15];
            #pragma unroll
            for (int v = 0; v < 8; ++v) acc[ct][v] = bv;
        }
        #pragma unroll
        for (int k = 0; k < 8; ++k) {
            const int c0 = 4*k + 2*half;
            float x0 = rp[c0], x1 = rp[c0 + 1];
            x0 = leaky(x0 * s_sc[c0]     + s_sh[c0]);
            x1 = leaky(x1 * s_sc[c0 + 1] + s_sh[c0 + 1]);
            v2f A; A.x = x0; A.y = x1;
            #pragma unroll
            for (int ct = 0; ct < NCT; ++ct) {
                const int o = ct * 16 + l15;
                v2f Bf; Bf.x = s_w[o*32 + c0]; Bf.y = s_w[o*32 + c0 + 1];
                acc[ct] = __builtin_amdgcn_wmma_f32_16x16x4_f32(false, A, false, Bf,
                                                                (short)0, acc[ct], false, false);
            }
        }
        const int rb = t * 16;
        #pragma unroll
        for (int ct = 0; ct < NCT; ++ct) {
            const int ch = ct * 16 + l15;
            #pragma unroll
            for (int v = 0; v < 8; ++v) {
                float d = acc[ct][v];
                bufOut[(size_t)(rb + v + 8*half) * OC + ch] = d;
                ps[ct]  += d;
                ps2[ct] += d * d;
            }
        }
    }
    #pragma unroll
    for (int ct = 0; ct < NCT; ++ct) {
        ps[ct]  += __shfl_xor(ps[ct],  16, 32);
        ps2[ct] += __shfl_xor(ps2[ct], 16, 32);
        if (half == 0) {
            s_red[wave][ct*16 + l15][0] = ps[ct];
            s_red[wave][ct*16 + l15][1] = ps2[ct];
        }
    }
    __syncthreads();
    if (threadIdx.x < OC * 2) {
        int ch = threadIdx.x >> 1, j = threadIdx.x & 1;
        float s = 0.f;
        #pragma unroll
        for (int w = 0; w < 8; ++w) s += s_red[w][ch][j];
        partial[(size_t)blockIdx.x * 128 + ch*2 + j] = s;
    }
}

// ---------------------------------------------------------------------------
// Deterministic final stat reduce: fixed-order sum over GRID_L partials,
// produces per-channel affine (scale = g*rsqrt(var+eps), shift = beta - mean*scale).
// ---------------------------------------------------------------------------
__global__ void stats_reduce_kernel(const float* __restrict__ partial,
                                    const float* __restrict__ g,
                                    const float* __restrict__ beta,
                                    float* __restrict__ nrm, int OC)
{
    const int ch = threadIdx.x;
    if (ch >= OC) return;
    float s = 0.f, s2 = 0.f;
    for (int w = 0; w < GRID_L; ++w) {
        s  += partial[(size_t)w * 128 + ch*2];
        s2 += partial[(size_t)w * 128 + ch*2 + 1];
    }
    const float invc = 1.0f / (float)ROWS;
    float mean = s * invc;
    float var  = s2 * invc - mean * mean;
    float istd = rsqrtf(var + EPS_);
    float sc   = g[ch] * istd;
    nrm[2*ch]     = sc;
    nrm[2*ch + 1] = beta[ch] - mean * sc;
}

// ---------------------------------------------------------------------------
// Finalize: norm2 + leaky + max over samples, write (B, 64, N).
// Consecutive threads cover consecutive channels -> fully coalesced reads.
// ---------------------------------------------------------------------------
__global__ void finalize_kernel(const float* __restrict__ bufIn, // ROWS x 64
                                const float* __restrict__ nrm,   // 64 x {scale,shift}
                                float* __restrict__ out)         // (B, 64, N)
{
    const int t  = blockIdx.x * 256 + threadIdx.x; // B*N*64 threads
    const int ch = t & 63;
    const int bn = t >> 6;
    const int n  = bn & (N_ - 1);
    const int b  = bn >> 12;
    const float sc = nrm[2*ch], sh = nrm[2*ch + 1];
    const float* p = bufIn + ((size_t)bn * NS) * 64 + ch;
    float m = -3.402823466e38f;
    #pragma unroll
    for (int s = 0; s < NS; ++s) {
        float x = p[(size_t)s * 64];
        x = leaky(x * sc + sh);
        m = fmaxf(m, x);
    }
    out[((size_t)(b * 64 + ch)) * N_ + n] = m;
}

// ---------------------------------------------------------------------------
extern "C" void kernel_launch(void* const* d_in, const int* in_sizes, int n_in,
                              void* d_out, int out_size, void* d_ws, size_t ws_size,
                              hipStream_t stream)
{
    (void)in_sizes; (void)n_in; (void)out_size; (void)ws_size;
    const float* points = (const float*)d_in[0];
    const float* w0  = (const float*)d_in[1];
    const float* b0  = (const float*)d_in[2];
    const float* g0  = (const float*)d_in[3];
    const float* be0 = (const float*)d_in[4];
    const float* w1  = (const float*)d_in[5];
    const float* b1  = (const float*)d_in[6];
    const float* g1  = (const float*)d_in[7];
    const float* be1 = (const float*)d_in[8];
    const float* w2  = (const float*)d_in[9];
    const float* b2  = (const float*)d_in[10];
    const float* g2  = (const float*)d_in[11];
    const float* be2 = (const float*)d_in[12];

    char* ws = (char*)d_ws;
    int*   idx     = (int*)  (ws + 0);                       //  2 MiB
    float* partial = (float*)(ws + ((size_t)4  << 20));      //  256 KiB
    float* nrm0    = (float*)(ws + ((size_t)8  << 20));
    float* nrm1    = (float*)(ws + ((size_t)8  << 20) + 4096);
    float* nrm2    = (float*)(ws + ((size_t)8  << 20) + 8192);
    float* bufA    = (float*)(ws + ((size_t)16 << 20));      // 64 MiB (32ch)
    float* bufB    = (float*)(ws + ((size_t)80 << 20));      // 128 MiB (64ch)
    float* out     = (float*)d_out;

    ball_query_kernel<<<B_ * (N_/256), 256, 0, stream>>>(points, idx);
    layer0_kernel<<<GRID_L, 256, 0, stream>>>(points, idx, w0, b0, bufA, partial);
    stats_reduce_kernel<<<1, 64, 0, stream>>>(partial, g0, be0, nrm0, 32);
    layer_ic32_kernel<32><<<GRID_L, 256, 0, stream>>>(bufA, bufA, w1, b1, nrm0, partial);
    stats_reduce_kernel<<<1, 64, 0, stream>>>(partial, g1, be1, nrm1, 32);
    layer_ic32_kernel<64><<<GRID_L, 256, 0, stream>>>(bufA, bufB, w2, b2, nrm1, partial);
    stats_reduce_kernel<<<1, 64, 0, stream>>>(partial, g2, be2, nrm2, 64);
    finalize_kernel<<<(B_*N_*64)/256, 256, 0, stream>>>(bufB, nrm2, out);
}